// MERG_22204980920675
// MI455X (gfx1250) — compile-verified
//
#include <hip/hip_runtime.h>
#include <hip/hip_bf16.h>

// ---------------------------------------------------------------------------
// MERG block for MI455X (gfx1250, wave32). All GEMMs run on the matrix pipe
// via V_WMMA_F32_16X16X4_F32 (fp32 in/out, no precision loss vs reference).
// Exact algebraic simplifications:
//  * encoder pass 2 over N*N rows collapses to N rows (linear attention sees
//    K/V only through KV = N*Km^T Vm and Ksum = N*colsum(Km)); then
//    index_edge[m] = out2[dst[m]].
//  * GatedGCN's returned e is discarded by the caller -> skip bne BN branch.
// Big-GEMM path (rows=16384, K=128): A strip staged in LDS (b128 loads,
// padded stride), 4 row-tiles per wave to reuse B fragments.
// ---------------------------------------------------------------------------

#define NNODE 512
#define DDIM  128
#define MEDGE 16384
#define SAPAD 140   // LDS row stride (floats): 140*4 % 16 == 0, avoids bank clash

typedef __attribute__((ext_vector_type(2))) float v2f;
typedef __attribute__((ext_vector_type(8))) float v8f;

enum { ACT_NONE = 0, ACT_RELU = 1, ACT_ELU1 = 2 };

// ----------------------------- WMMA GEMM -----------------------------------
// C[rows,ncols] = act(alpha*op(A)@B *rowscale + bias) (+ gadd[gidx[row]] row)
// grid.x = rows/(16*RPW), 256 threads (8 waves); wave owns 16x16 col tiles.
// RPW==4 requires K==128 and !TRANSA (A strip staged in LDS).
template <int ACT, bool TRANSA, bool HB, bool HRS, bool HG, int RPW>
__global__ __launch_bounds__(256) void gemm_wmma(
    const float* __restrict__ A, int lda,
    const float* __restrict__ B, int ldb,
    const float* __restrict__ bias,
    const float* __restrict__ rowscale,
    const float* __restrict__ gadd, const int* __restrict__ gidx,
    float* __restrict__ C, int ldc, int K, int ncols, float alpha) {
  __shared__ float sA[(RPW == 4) ? 64 * SAPAD : 1];
  const int lane  = threadIdx.x & 31;
  const int wave  = threadIdx.x >> 5;
  const int row0  = blockIdx.x * (16 * RPW);
  const int l15   = lane & 15;
  const int khalf = (lane >> 4) << 1;  // lanes 0-15: K0/K1, lanes 16-31: K2/K3
  const int ntile = ncols >> 4;

  if (RPW == 4) {  // cooperative stage of 64 x 128 A strip (K==128)
    for (int i = threadIdx.x; i < 64 * 32; i += 256) {
      const int row = i >> 5;
      const int c4  = (i & 31) << 2;
      const float4 v = *(const float4*)(A + (size_t)(row0 + row) * lda + c4);
      *(float4*)(&sA[row * SAPAD + c4]) = v;
    }
    __syncthreads();
  }

  for (int nt = wave; nt < ntile; nt += 8) {
    const int col0 = nt << 4;
    v8f acc[RPW];
#pragma unroll
    for (int t = 0; t < RPW; ++t) acc[t] = (v8f){0.f,0.f,0.f,0.f,0.f,0.f,0.f,0.f};
    const float* Bp = B + (size_t)khalf * ldb + col0 + l15;
#pragma unroll 4
    for (int k0 = 0; k0 < K; k0 += 4) {
      v2f b;
      b.x = Bp[0];
      b.y = Bp[ldb];
      Bp += (size_t)4 * ldb;
#pragma unroll
      for (int t = 0; t < RPW; ++t) {
        v2f a;
        if (RPW == 4) {
          a = *(const v2f*)(&sA[(16 * t + l15) * SAPAD + k0 + khalf]);
        } else if (TRANSA) {  // logical A[r][k] = A[k*lda + r]
          a.x = A[(size_t)(k0 + khalf) * lda + row0 + 16 * t + l15];
          a.y = A[(size_t)(k0 + khalf + 1) * lda + row0 + 16 * t + l15];
        } else {
          a = *(const v2f*)(A + (size_t)(row0 + 16 * t + l15) * lda + k0 + khalf);
        }
        acc[t] = __builtin_amdgcn_wmma_f32_16x16x4_f32(false, a, false, b,
                                                       (short)0, acc[t], false,
                                                       false);
      }
    }
    const int col  = col0 + l15;
    const int rsub = (lane >> 4) << 3;  // C/D layout: vgpr i -> M=i (+8 hi half)
    const float bv = HB ? bias[col] : 0.0f;
#pragma unroll
    for (int t = 0; t < RPW; ++t) {
      const int rbase = row0 + 16 * t + rsub;
#pragma unroll
      for (int i = 0; i < 8; ++i) {
        const int r = rbase + i;
        float v = acc[t][i] * alpha;
        if (HRS) v *= rowscale[r];
        v += bv;
        if (ACT == ACT_RELU) v = fmaxf(v, 0.0f);
        if (ACT == ACT_ELU1) v = v > 0.0f ? v + 1.0f : __expf(v);
        if (HG) v += gadd[(size_t)gidx[r] * ncols + col];
        C[(size_t)r * ldc + col] = v;
      }
    }
  }
}

// --------------------------- small helper kernels ---------------------------
__global__ void zero_kernel(float* p, int n) {
  int i = blockIdx.x * blockDim.x + threadIdx.x;
  if (i < n) p[i] = 0.0f;
}

__global__ __launch_bounds__(256) void gcn_edge_kernel(
    const float* __restrict__ Bh, const float* __restrict__ Dh,
    const float* __restrict__ Eh, const float* __restrict__ Ce,
    const int* __restrict__ src, const int* __restrict__ dst,
    float* __restrict__ num, float* __restrict__ den) {
  int idx = blockIdx.x * blockDim.x + threadIdx.x;
  int m = idx >> 7, d = idx & 127;
  if (m >= MEDGE) return;
  int s = src[m], t = dst[m];
  float en = Dh[s * DDIM + d] + Eh[t * DDIM + d] + Ce[(size_t)m * DDIM + d];
  float sg = 1.0f / (1.0f + __expf(-en));
  atomicAdd(&num[t * DDIM + d], Bh[s * DDIM + d] * sg);
  atomicAdd(&den[t * DDIM + d], sg);
}

__global__ void hnew_kernel(const float* Ah, const float* num, const float* den,
                            float* hn, int n) {
  int i = blockIdx.x * blockDim.x + threadIdx.x;
  if (i < n) hn[i] = Ah[i] + num[i] / (den[i] + 1e-6f);
}

// per-column batch stats (BatchNorm1d, training-mode stats)
__global__ __launch_bounds__(256) void colstats_kernel(const float* __restrict__ X,
                                                       int R, float* mu,
                                                       float* rstd) {
  __shared__ float ss[256], sq[256];
  const int c = blockIdx.x;
  float s = 0.f, q = 0.f;
  for (int r = threadIdx.x; r < R; r += 256) {
    float v = X[(size_t)r * DDIM + c];
    s += v;
    q += v * v;
  }
  ss[threadIdx.x] = s;
  sq[threadIdx.x] = q;
  __syncthreads();
  for (int st = 128; st > 0; st >>= 1) {
    if (threadIdx.x < st) {
      ss[threadIdx.x] += ss[threadIdx.x + st];
      sq[threadIdx.x] += sq[threadIdx.x + st];
    }
    __syncthreads();
  }
  if (threadIdx.x == 0) {
    float m = ss[0] / R;
    float var = sq[0] / R - m * m;
    mu[c] = m;
    rstd[c] = rsqrtf(var + 1e-5f);
  }
}

__global__ void bn_apply_kernel(const float* X, const float* res, const float* g,
                                const float* b, const float* mu,
                                const float* rstd, float* out, int total) {
  int i = blockIdx.x * blockDim.x + threadIdx.x;
  if (i >= total) return;
  int d = i & 127;
  float v = fmaxf(g[d] * (X[i] - mu[d]) * rstd[d] + b[d], 0.0f);
  out[i] = (res ? res[i] : 0.0f) + v;
}

__global__ __launch_bounds__(128) void colsum_kernel(const float* X, int R,
                                                     float scale, float* out) {
  int c = threadIdx.x;
  float s = 0.f;
  for (int r = 0; r < R; ++r) s += X[(size_t)r * DDIM + c];
  out[c] = s * scale;
}

__global__ __launch_bounds__(128) void gemv_kernel(const float* x, const float* W,
                                                   float* out, int K, int act) {
  int j = threadIdx.x;
  float s = 0.f;
  for (int d = 0; d < K; ++d) s += x[d] * W[(size_t)d * DDIM + j];
  if (act == ACT_ELU1) s = s > 0.f ? s + 1.f : __expf(s);
  out[j] = s;
}

// encoder pass 1: src_fea rows all equal global_g -> msg[i] = t/(t+1e-6)*vrow
__global__ __launch_bounds__(128) void msg_uniform_kernel(
    const float* __restrict__ Q, const float* __restrict__ kvec,
    const float* __restrict__ vvec, float* __restrict__ msg) {
  __shared__ float red[128];
  int i = blockIdx.x, d = threadIdx.x;
  red[d] = Q[(size_t)i * DDIM + d] * kvec[d];
  __syncthreads();
  for (int st = 64; st > 0; st >>= 1) {
    if (d < st) red[d] += red[d + st];
    __syncthreads();
  }
  float t = (float)NNODE * red[0];
  msg[(size_t)i * DDIM + d] = (t / (t + 1e-6f)) * vvec[d];
}

__global__ __launch_bounds__(128) void rowz_kernel(const float* __restrict__ Q,
                                                   const float* __restrict__ ksum,
                                                   float* __restrict__ Z) {
  __shared__ float red[128];
  int i = blockIdx.x, d = threadIdx.x;
  red[d] = Q[(size_t)i * DDIM + d] * ksum[d];
  __syncthreads();
  for (int st = 64; st > 0; st >>= 1) {
    if (d < st) red[d] += red[d + st];
    __syncthreads();
  }
  if (d == 0) Z[i] = 1.0f / (red[0] + 1e-6f);
}

__global__ __launch_bounds__(128) void ln_kernel(const float* X, const float* res,
                                                 const float* g, const float* b,
                                                 float* out) {
  __shared__ float red[128];
  int i = blockIdx.x, d = threadIdx.x;
  float v = X[(size_t)i * DDIM + d];
  red[d] = v;
  __syncthreads();
  for (int st = 64; st > 0; st >>= 1) {
    if (d < st) red[d] += red[d + st];
    __syncthreads();
  }
  float mu = red[0] * (1.0f / 128.0f);
  __syncthreads();
  float c = v - mu;
  red[d] = c * c;
  __syncthreads();
  for (int st = 64; st > 0; st >>= 1) {
    if (d < st) red[d] += red[d + st];
    __syncthreads();
  }
  float o = g[d] * c * rsqrtf(red[0] * (1.0f / 128.0f) + 1e-5f) + b[d];
  out[(size_t)i * DDIM + d] = (res ? res[(size_t)i * DDIM + d] : 0.0f) + o;
}

__global__ __launch_bounds__(128) void copycols_kernel(const float* src, float* dst,
                                                       int lddst, int coloff) {
  int i = blockIdx.x, d = threadIdx.x;
  dst[(size_t)i * lddst + coloff + d] = src[(size_t)i * DDIM + d];
}

// Conv1d(in=2,out=1,k=3,pad=1) over stacked [h[src], h[dst]] rows
__global__ __launch_bounds__(128) void conv_edge_kernel(
    const float* __restrict__ h, const int* __restrict__ src,
    const int* __restrict__ dst, const float* __restrict__ cw,
    const float* __restrict__ cb, float* __restrict__ loc) {
  __shared__ float s0[130], s1[130];
  int m = blockIdx.x, d = threadIdx.x;
  s0[d + 1] = h[(size_t)src[m] * DDIM + d];
  s1[d + 1] = h[(size_t)dst[m] * DDIM + d];
  if (d == 0) { s0[0] = s1[0] = s0[129] = s1[129] = 0.0f; }
  __syncthreads();
  float v = cw[0] * s0[d] + cw[1] * s0[d + 1] + cw[2] * s0[d + 2] +
            cw[3] * s1[d] + cw[4] * s1[d + 1] + cw[5] * s1[d + 2] + cb[0];
  loc[(size_t)m * DDIM + d] = v;
}

// ------------------------------- host side ----------------------------------
static void launch_gemm(hipStream_t st, const float* A, int lda, const float* B,
                        int ldb, const float* bias, const float* rs,
                        const float* gadd, const int* gidx, float* C, int ldc,
                        int rows, int K, int ncols, int act, bool transA,
                        float alpha) {
  dim3 b(256);
  const bool big = (rows % 64 == 0) && rows >= 1024 && K == 128 && !transA;
  if (transA) {
    gemm_wmma<ACT_NONE, true, false, false, false, 1><<<rows / 16, b, 0, st>>>(A, lda, B, ldb, bias, rs, gadd, gidx, C, ldc, K, ncols, alpha);
  } else if (gadd) {
    gemm_wmma<ACT_NONE, false, true, false, true, 4><<<rows / 64, b, 0, st>>>(A, lda, B, ldb, bias, rs, gadd, gidx, C, ldc, K, ncols, alpha);
  } else if (rs) {
    gemm_wmma<ACT_NONE, false, false, true, false, 1><<<rows / 16, b, 0, st>>>(A, lda, B, ldb, bias, rs, gadd, gidx, C, ldc, K, ncols, alpha);
  } else if (act == ACT_ELU1) {
    gemm_wmma<ACT_ELU1, false, false, false, false, 1><<<rows / 16, b, 0, st>>>(A, lda, B, ldb, bias, rs, gadd, gidx, C, ldc, K, ncols, alpha);
  } else if (act == ACT_RELU) {
    gemm_wmma<ACT_RELU, false, true, false, false, 1><<<rows / 16, b, 0, st>>>(A, lda, B, ldb, bias, rs, gadd, gidx, C, ldc, K, ncols, alpha);
  } else if (bias && big) {
    gemm_wmma<ACT_NONE, false, true, false, false, 4><<<rows / 64, b, 0, st>>>(A, lda, B, ldb, bias, rs, gadd, gidx, C, ldc, K, ncols, alpha);
  } else if (bias) {
    gemm_wmma<ACT_NONE, false, true, false, false, 1><<<rows / 16, b, 0, st>>>(A, lda, B, ldb, bias, rs, gadd, gidx, C, ldc, K, ncols, alpha);
  } else {
    gemm_wmma<ACT_NONE, false, false, false, false, 1><<<rows / 16, b, 0, st>>>(A, lda, B, ldb, bias, rs, gadd, gidx, C, ldc, K, ncols, alpha);
  }
}

// Input index map: top-level dict insertion order (h, e, src, dst), then
// params flattened as a JAX pytree (dict keys sorted, list order preserved).
enum {
  IDX_H = 0, IDX_E, IDX_SRC, IDX_DST,
  IDX_W2, IDX_W4, IDX_B2, IDX_B4, IDX_BN_B, IDX_BN_G, IDX_CONV_B, IDX_CONV_W,
  IDX_ENC_W1, IDX_ENC_W2, IDX_ENC_WK, IDX_ENC_WM, IDX_ENC_WQ, IDX_ENC_WV,
  IDX_ENC_B1, IDX_ENC_B2, IDX_ENC_BM, IDX_ENC_LN1B, IDX_ENC_LN1G,
  IDX_ENC_LN2B, IDX_ENC_LN2G,
  IDX_GCN0 = 25, IDX_GCN1 = 39
};
// gcn dict sorted-key offsets
enum { G_AB = 0, G_AW, G_BB, G_BW, G_CB, G_CW, G_DB, G_DW, G_EB, G_EW,
       G_BNEB, G_BNEG, G_BNHB, G_BNHG };

extern "C" void kernel_launch(void* const* d_in, const int* in_sizes, int n_in,
                              void* d_out, int out_size, void* d_ws,
                              size_t ws_size, hipStream_t stream) {
  (void)in_sizes; (void)n_in; (void)out_size; (void)ws_size;
  const float* h   = (const float*)d_in[IDX_H];
  const float* e   = (const float*)d_in[IDX_E];
  const int* src   = (const int*)d_in[IDX_SRC];
  const int* dst   = (const int*)d_in[IDX_DST];
  const float* Wq  = (const float*)d_in[IDX_ENC_WQ];
  const float* Wk  = (const float*)d_in[IDX_ENC_WK];
  const float* Wv  = (const float*)d_in[IDX_ENC_WV];
  const float* Wm  = (const float*)d_in[IDX_ENC_WM];
  const float* bm  = (const float*)d_in[IDX_ENC_BM];
  const float* eW1 = (const float*)d_in[IDX_ENC_W1];
  const float* eb1 = (const float*)d_in[IDX_ENC_B1];
  const float* eW2 = (const float*)d_in[IDX_ENC_W2];
  const float* eb2 = (const float*)d_in[IDX_ENC_B2];
  float* ws = (float*)d_ws;

  // workspace carve-up (floats)
  size_t off = 0;
  auto A = [&](size_t n) { float* p = ws + off; off += n; return p; };
  float* big0  = A((size_t)MEDGE * DDIM);  // Ce, then conv loc
  float* hA    = A(NNODE * DDIM);
  float* hB    = A(NNODE * DDIM);
  float* Ah    = A(NNODE * DDIM);
  float* Bh    = A(NNODE * DDIM);
  float* Dh    = A(NNODE * DDIM);
  float* Eh    = A(NNODE * DDIM);
  float* numb  = A(NNODE * DDIM);
  float* denb  = A(NNODE * DDIM);
  float* hn    = A(NNODE * DDIM);
  float* bufQ  = A(NNODE * DDIM);
  float* bufK  = A(NNODE * DDIM);
  float* bufV  = A(NNODE * DDIM);
  float* bufMs = A(NNODE * DDIM);
  float* bufM  = A(NNODE * DDIM);
  float* bufL  = A(NNODE * DDIM);
  float* bufM2 = A(NNODE * DDIM);
  float* qfea  = A(NNODE * DDIM);
  float* out2  = A(NNODE * DDIM);
  float* gnode = A(NNODE * DDIM);
  float* catb  = A(NNODE * 2 * DDIM);
  float* bufT  = A(NNODE * 2 * DDIM);
  float* bufKV = A(DDIM * DDIM);
  float* gvec  = A(DDIM);
  float* kvec  = A(DDIM);
  float* vvec  = A(DDIM);
  float* zbuf  = A(NNODE);
  float* mu    = A(DDIM);
  float* rstd  = A(DDIM);

  const int ND = NNODE * DDIM, MD = MEDGE * DDIM;

  // ---------------- GatedGCN stack (L=2), e-branch BN is dead code ----------
  const float* hin = h;
  float* houts[2] = {hA, hB};
  for (int l = 0; l < 2; ++l) {
    const int base = (l == 0) ? IDX_GCN0 : IDX_GCN1;
    const float** P = (const float**)(d_in + base);
    launch_gemm(stream, hin, DDIM, P[G_AW], DDIM, P[G_AB], 0, 0, 0, Ah, DDIM, NNODE, DDIM, DDIM, ACT_NONE, false, 1.f);
    launch_gemm(stream, hin, DDIM, P[G_BW], DDIM, P[G_BB], 0, 0, 0, Bh, DDIM, NNODE, DDIM, DDIM, ACT_NONE, false, 1.f);
    launch_gemm(stream, hin, DDIM, P[G_DW], DDIM, P[G_DB], 0, 0, 0, Dh, DDIM, NNODE, DDIM, DDIM, ACT_NONE, false, 1.f);
    launch_gemm(stream, hin, DDIM, P[G_EW], DDIM, P[G_EB], 0, 0, 0, Eh, DDIM, NNODE, DDIM, DDIM, ACT_NONE, false, 1.f);
    launch_gemm(stream, e, DDIM, P[G_CW], DDIM, P[G_CB], 0, 0, 0, big0, DDIM, MEDGE, DDIM, DDIM, ACT_NONE, false, 1.f);
    zero_kernel<<<(ND + 255) / 256, 256, 0, stream>>>(numb, ND);
    zero_kernel<<<(ND + 255) / 256, 256, 0, stream>>>(denb, ND);
    gcn_edge_kernel<<<MD / 256, 256, 0, stream>>>(Bh, Dh, Eh, big0, src, dst, numb, denb);
    hnew_kernel<<<ND / 256, 256, 0, stream>>>(Ah, numb, denb, hn, ND);
    colstats_kernel<<<DDIM, 256, 0, stream>>>(hn, NNODE, mu, rstd);
    bn_apply_kernel<<<ND / 256, 256, 0, stream>>>(hn, hin, P[G_BNHG], P[G_BNHB], mu, rstd, houts[l], ND);
    hin = houts[l];
  }
  // global mean over nodes
  colsum_kernel<<<1, 128, 0, stream>>>(hB, NNODE, 1.0f / NNODE, gvec);

  // ---------------- encoder pass 1: x = h, src_fea = broadcast(global_g) ----
  launch_gemm(stream, h, DDIM, Wq, DDIM, 0, 0, 0, 0, bufQ, DDIM, NNODE, DDIM, DDIM, ACT_ELU1, false, 1.f);
  gemv_kernel<<<1, 128, 0, stream>>>(gvec, Wk, kvec, DDIM, ACT_ELU1);
  gemv_kernel<<<1, 128, 0, stream>>>(gvec, Wv, vvec, DDIM, ACT_NONE);
  msg_uniform_kernel<<<NNODE, 128, 0, stream>>>(bufQ, kvec, vvec, bufMs);
  launch_gemm(stream, bufMs, DDIM, Wm, DDIM, bm, 0, 0, 0, bufM, DDIM, NNODE, DDIM, DDIM, ACT_NONE, false, 1.f);
  ln_kernel<<<NNODE, 128, 0, stream>>>(bufM, 0, (const float*)d_in[IDX_ENC_LN1G], (const float*)d_in[IDX_ENC_LN1B], bufL);
  copycols_kernel<<<NNODE, 128, 0, stream>>>(h, catb, 2 * DDIM, 0);
  copycols_kernel<<<NNODE, 128, 0, stream>>>(bufL, catb, 2 * DDIM, DDIM);
  launch_gemm(stream, catb, 2 * DDIM, eW1, 2 * DDIM, eb1, 0, 0, 0, bufT, 2 * DDIM, NNODE, 2 * DDIM, 2 * DDIM, ACT_RELU, false, 1.f);
  launch_gemm(stream, bufT, 2 * DDIM, eW2, DDIM, eb2, 0, 0, 0, bufM2, DDIM, NNODE, 2 * DDIM, DDIM, ACT_NONE, false, 1.f);
  ln_kernel<<<NNODE, 128, 0, stream>>>(bufM2, h, (const float*)d_in[IDX_ENC_LN2G], (const float*)d_in[IDX_ENC_LN2B], qfea);

  // ---------------- encoder pass 2 (collapsed N*N -> N) ---------------------
  launch_gemm(stream, qfea, DDIM, Wq, DDIM, 0, 0, 0, 0, bufQ, DDIM, NNODE, DDIM, DDIM, ACT_ELU1, false, 1.f);
  launch_gemm(stream, qfea, DDIM, Wk, DDIM, 0, 0, 0, 0, bufK, DDIM, NNODE, DDIM, DDIM, ACT_ELU1, false, 1.f);
  launch_gemm(stream, qfea, DDIM, Wv, DDIM, 0, 0, 0, 0, bufV, DDIM, NNODE, DDIM, DDIM, ACT_NONE, false, 1.f);
  // KV = N * Km^T @ Vm   (A transposed, alpha = N)
  launch_gemm(stream, bufK, DDIM, bufV, DDIM, 0, 0, 0, 0, bufKV, DDIM, DDIM, NNODE, DDIM, ACT_NONE, true, (float)NNODE);
  colsum_kernel<<<1, 128, 0, stream>>>(bufK, NNODE, (float)NNODE, kvec);  // Ksum
  rowz_kernel<<<NNODE, 128, 0, stream>>>(bufQ, kvec, zbuf);
  launch_gemm(stream, bufQ, DDIM, bufKV, DDIM, 0, zbuf, 0, 0, bufMs, DDIM, NNODE, DDIM, DDIM, ACT_NONE, false, 1.f);
  launch_gemm(stream, bufMs, DDIM, Wm, DDIM, bm, 0, 0, 0, bufM, DDIM, NNODE, DDIM, DDIM, ACT_NONE, false, 1.f);
  ln_kernel<<<NNODE, 128, 0, stream>>>(bufM, 0, (const float*)d_in[IDX_ENC_LN1G], (const float*)d_in[IDX_ENC_LN1B], bufL);
  copycols_kernel<<<NNODE, 128, 0, stream>>>(qfea, catb, 2 * DDIM, 0);
  copycols_kernel<<<NNODE, 128, 0, stream>>>(bufL, catb, 2 * DDIM, DDIM);
  launch_gemm(stream, catb, 2 * DDIM, eW1, 2 * DDIM, eb1, 0, 0, 0, bufT, 2 * DDIM, NNODE, 2 * DDIM, 2 * DDIM, ACT_RELU, false, 1.f);
  launch_gemm(stream, bufT, 2 * DDIM, eW2, DDIM, eb2, 0, 0, 0, bufM2, DDIM, NNODE, 2 * DDIM, DDIM, ACT_NONE, false, 1.f);
  ln_kernel<<<NNODE, 128, 0, stream>>>(bufM2, qfea, (const float*)d_in[IDX_ENC_LN2G], (const float*)d_in[IDX_ENC_LN2B], out2);

  // per-node global edge features: gnode = out2 @ W4 + b4
  launch_gemm(stream, out2, DDIM, (const float*)d_in[IDX_W4], DDIM, (const float*)d_in[IDX_B4], 0, 0, 0, gnode, DDIM, NNODE, DDIM, DDIM, ACT_NONE, false, 1.f);

  // ---------------- local branch + fuse + final BN/ReLU ---------------------
  conv_edge_kernel<<<MEDGE, 128, 0, stream>>>(h, src, dst, (const float*)d_in[IDX_CONV_W], (const float*)d_in[IDX_CONV_B], big0);
  float* outp = (float*)d_out;
  // out = loc @ W2 + b2 + gnode[dst]   (pre-BN)
  launch_gemm(stream, big0, DDIM, (const float*)d_in[IDX_W2], DDIM, (const float*)d_in[IDX_B2], 0, gnode, dst, outp, DDIM, MEDGE, DDIM, DDIM, ACT_NONE, false, 1.f);
  colstats_kernel<<<DDIM, 256, 0, stream>>>(outp, MEDGE, mu, rstd);
  bn_apply_kernel<<<MD / 256, 256, 0, stream>>>(outp, 0, (const float*)d_in[IDX_BN_G], (const float*)d_in[IDX_BN_B], mu, rstd, outp, MD);
}